// BorderAlign_23845658427885
// MI455X (gfx1250) — compile-verified
//
#include <hip/hip_runtime.h>
#include <math.h>

// BorderAlign for fixed problem: N=2, C4=128 (C=32), H=W=100, POOL=10.
// One wave32 per (n, box, border); lane == channel (exactly fills a wave).
#define BA_N    2
#define BA_C    32
#define BA_H    100
#define BA_W    100
#define BA_HW   (BA_H * BA_W)
#define BA_POOL 10
#define BA_P1   (BA_POOL + 1)
#define WAVES_PER_BLOCK 8
#define BA_ITEMS  (BA_N * BA_HW * 4)            // 80000 (n,box,border) items
#define BA_BLOCKS (BA_ITEMS / WAVES_PER_BLOCK)  // 10000

#if __has_builtin(__builtin_amdgcn_global_load_async_to_lds_b128) && \
    __has_builtin(__builtin_amdgcn_s_wait_asynccnt)
#define USE_ASYNC_LDS 1
#else
#define USE_ASYNC_LDS 0
#endif

#if USE_ASYNC_LDS
// b128 async builtin takes v4i pointers in AS1 (global) / AS3 (LDS).
typedef int v4i __attribute__((ext_vector_type(4)));
typedef __attribute__((address_space(1))) v4i* gv4i_p;
typedef __attribute__((address_space(3))) v4i* lv4i_p;
#endif

// 8-byte pair with only 4-byte alignment guarantee (xl may be odd):
// default DWORD alignment mode permits b64 loads at 4B alignment.
struct __attribute__((packed, aligned(4))) F2 { float a, b; };

__global__ __launch_bounds__(256) void border_align_kernel(
    const float* __restrict__ feat,    // [N, 4*C, H, W]
    const float* __restrict__ boxes,   // [N, HW, 4] (x1,y1,x2,y2)
    float* __restrict__ out) {         // [N, C, HW, 4]
  const int lane = threadIdx.x & 31;   // channel within border group
  const int wave = threadIdx.x >> 5;
  const int item = blockIdx.x * WAVES_PER_BLOCK + wave;  // wave-uniform
  const int border = item & 3;         // 0=top,1=left,2=bottom,3=right
  const int boxId  = item >> 2;        // == n*HW + hw
  const int hw = boxId % BA_HW;
  const int n  = boxId / BA_HW;

  const float* boxp = boxes + (size_t)boxId * 4;  // 16B aligned

  float bx1, by1, bx2, by2;
#if USE_ASYNC_LDS
  // Stage the 16B box through the CDNA5 async-to-LDS path; ASYNCcnt is
  // per-wave so each wave only waits on its own transfer (no block barrier).
  __shared__ __attribute__((aligned(16))) float sbox[WAVES_PER_BLOCK * 4];
  if (lane == 0) {
    __builtin_amdgcn_global_load_async_to_lds_b128(
        (gv4i_p)boxp, (lv4i_p)&sbox[wave * 4],
        /*offset=*/0, /*cpol=*/0);
  }
  __builtin_amdgcn_s_wait_asynccnt(0);
  asm volatile("" ::: "memory");  // LDS now holds the box; block reordering
  {
    const float4 b = *(const float4*)&sbox[wave * 4];  // one ds_load_b128
    bx1 = b.x; by1 = b.y; bx2 = b.z; by2 = b.w;
  }
#else
  bx1 = boxp[0]; by1 = boxp[1]; bx2 = boxp[2]; by2 = boxp[3];
#endif

  const float bw = bx2 - bx1;
  const float bh = by2 - by1;

  // Border path p(t) = (xb + cx*t, yb + cy*t), t = s/POOL. Wave-uniform.
  float xb, yb, cx, cy;
  switch (border) {
    case 0:  xb = bx1; yb = by1; cx = bw;   cy = 0.0f; break;  // top
    case 1:  xb = bx1; yb = by1; cx = 0.0f; cy = bh;   break;  // left
    case 2:  xb = bx2; yb = by2; cx = -bw;  cy = 0.0f; break;  // bottom
    default: xb = bx2; yb = by2; cx = 0.0f; cy = -bh;  break;  // right
  }

  // Channel plane for this lane: feat[n][border*C + lane][.][.]
  const float* __restrict__ fp =
      feat + (size_t)((n * 4 + border) * BA_C + lane) * BA_HW;

  float acc = -INFINITY;
#pragma unroll
  for (int s = 0; s < BA_P1; ++s) {
    const float t = (float)s / (float)BA_POOL;  // folds to exact constant
    const float x = xb + cx * t;
    const float y = yb + cy * t;
    const bool valid = (y > -1.0f) && (y < (float)BA_H) &&
                       (x > -1.0f) && (x < (float)BA_W);
    const float xc = fminf(fmaxf(x, 0.0f), (float)(BA_W - 1));
    const float yc = fminf(fmaxf(y, 0.0f), (float)(BA_H - 1));
    int xl = (int)xc; if (xl > BA_W - 2) xl = BA_W - 2;  // xc>=0: trunc==floor
    int yl = (int)yc; if (yl > BA_H - 2) yl = BA_H - 2;
    const float lx = xc - (float)xl;
    const float ly = yc - (float)yl;
    const float hx = 1.0f - lx;
    const float hy = 1.0f - ly;
    const float* p0 = fp + (yl * BA_W + xl);
    const F2 r0 = *(const F2*)p0;            // (yl ,xl),(yl ,xl+1): 1x b64
    const F2 r1 = *(const F2*)(p0 + BA_W);   // (yl+1,xl),(yl+1,xl+1): 1x b64
    const float v = hy * (hx * r0.a + lx * r0.b) +
                    ly * (hx * r1.a + lx * r1.b);
    acc = fmaxf(acc, valid ? v : 0.0f);      // invalid samples contribute 0
  }

  // out[n][c=lane][hw][border]
  out[((size_t)(n * BA_C + lane) * BA_HW + hw) * 4 + border] = acc;
}

extern "C" void kernel_launch(void* const* d_in, const int* in_sizes, int n_in,
                              void* d_out, int out_size, void* d_ws, size_t ws_size,
                              hipStream_t stream) {
  const float* feat  = (const float*)d_in[0];  // [2,128,100,100] f32
  const float* boxes = (const float*)d_in[1];  // [2,10000,4] f32
  float* out = (float*)d_out;                  // [2,32,10000,4] f32
  (void)in_sizes; (void)n_in; (void)out_size; (void)d_ws; (void)ws_size;
  hipLaunchKernelGGL(border_align_kernel, dim3(BA_BLOCKS), dim3(256), 0, stream,
                     feat, boxes, out);
}